// DynamicGC_21792664060191
// MI455X (gfx1250) — compile-verified
//
#include <hip/hip_runtime.h>
#include <hip/hip_bf16.h>

// ---------------------------------------------------------------------------
// DynamicGC on MI455X (gfx1250): chained FP32 WMMA GEMMs + masked softmax.
// One workgroup (256 thr = 8 wave32) per (batch b, head h).
// feat/node GEMMs (K=208) are K-split across wave pairs so all 8 waves drive
// the matrix pipe; partials reduced through LDS.
// ---------------------------------------------------------------------------

typedef __attribute__((ext_vector_type(2))) float v2f;
typedef __attribute__((ext_vector_type(8))) float v8f;

#define NV    199     // graph nodes
#define NPAD  208     // 13 * 16
#define KHALF (NPAD / 2)   // 104, multiple of 4
#define FDIM  64
#define UDIM  64
#define HNUM  4
#define OUTC  (HNUM * UDIM + NV)   // 455
#define NEGF  1e16f

// LDS partition (floats)
#define OFF_XS   0                         // [208][64]  X_b
#define OFF_KS   (OFF_XS  + NPAD * FDIM)   // [64][208]  kernels_h
#define OFF_FC   (OFF_KS  + FDIM * NPAD)   // [64][64]   fc_h
#define OFF_AW   (OFF_FC  + FDIM * UDIM)   // [16][208]  AWT block / A block / node partials
#define OFF_FS   (OFF_AW  + 16 * NPAD)     // [16][64]   feat block
#define OFF_DS   (OFF_FS  + 16 * FDIM)     // [16][208]  feat partials / dense / mask
#define OFF_NS   (OFF_DS  + 16 * NPAD)     // [16][64]   node block
#define OFF_B1   (OFF_NS  + 16 * FDIM)     // [208]      bias1_h
#define OFF_B2   (OFF_B1  + NPAD)          // [64]       bias2_h
#define SMEM_FLOATS (OFF_B2 + UDIM)
#define SMEM_BYTES  (SMEM_FLOATS * sizeof(float))

__device__ __forceinline__ v8f wmma_f32(v2f a, v2f b, v8f c) {
    return __builtin_amdgcn_wmma_f32_16x16x4_f32(
        /*neg_a=*/false, a, /*neg_b=*/false, b,
        /*c_mod=*/(short)0, c, /*reuse_a=*/false, /*reuse_b=*/false);
}

// 16x16 tile GEMM over K range [kbeg,kend) with f32 WMMA 16x16x4.
// A (16 x lda): lanes 0-15 rows M=0..15 {K=k,k+1}; lanes 16-31 {K=k+2,k+3}
// B (lda x 16): lanes 0-15 cols N=n0..n0+15 of rows {k,k+1}; lanes 16-31 {k+2,k+3}
__device__ __forceinline__ v8f gemm_tile(const float* __restrict__ As, int lda,
                                         const float* __restrict__ Bs, int ldb,
                                         int n0, int kbeg, int kend, v8f c) {
    const int lane = threadIdx.x & 31;
    const int m    = lane & 15;
    const int koff = (lane >> 4) << 1;          // 0 or 2
    const float* ap = As + m * lda + kbeg + koff;
    const float* bp = Bs + (kbeg + koff) * ldb + n0 + m;
    for (int k = kbeg; k < kend; k += 4) {
        v2f a; a.x = ap[0]; a.y = ap[1];
        v2f b; b.x = bp[0]; b.y = bp[ldb];
        c = wmma_f32(a, b, c);
        ap += 4;
        bp += 4 * ldb;
    }
    return c;
}

// ---------------------------------------------------------------------------
// Prep: AWT[h][m][n] = A[n][m] * params[h][n][m], zero-padded to 208x208.
// ---------------------------------------------------------------------------
__global__ void prep_awt(const float* __restrict__ A,
                         const float* __restrict__ params,
                         float* __restrict__ AWT) {
    int idx = blockIdx.x * 256 + threadIdx.x;
    const int total = HNUM * NPAD * NPAD;
    if (idx >= total) return;
    int h = idx / (NPAD * NPAD);
    int r = idx - h * NPAD * NPAD;
    int m = r / NPAD, n = r - m * NPAD;
    float v = 0.f;
    if (m < NV && n < NV)
        v = A[n * NV + m] * params[(h * NV + n) * NV + m];
    AWT[idx] = v;
}

// ---------------------------------------------------------------------------
// Main: one workgroup per (b, h).
// ---------------------------------------------------------------------------
__global__ __launch_bounds__(256)
void gc_main(const float* __restrict__ X,      // (B,199,64)
             const float* __restrict__ A,      // (199,199)
             const float* __restrict__ kernels,// (4,64,199)
             const float* __restrict__ fc,     // (4,64,64)
             const float* __restrict__ bias1,  // (4,199)
             const float* __restrict__ bias2,  // (4,64)
             const float* __restrict__ AWT,    // (4,208,208) workspace
             float* __restrict__ out) {        // (B,199,455)
    const int b    = blockIdx.x;
    const int h    = blockIdx.y;
    const int tid  = threadIdx.x;
    const int wave = tid >> 5;
    const int lane = tid & 31;
    const int hi   = lane >> 4;
    const int lo   = lane & 15;

    extern __shared__ float smem[];
    float* xs  = smem + OFF_XS;
    float* ks  = smem + OFF_KS;
    float* fcs = smem + OFF_FC;
    float* awt = smem + OFF_AW;   // also: adjacency block, node partials
    float* fs  = smem + OFF_FS;
    float* ds  = smem + OFF_DS;   // also: feat partials
    float* ns  = smem + OFF_NS;
    float* b1s = smem + OFF_B1;
    float* b2s = smem + OFF_B2;

    // ---- stage per-(b,h) constants into LDS --------------------------------
    for (int i = tid; i < NPAD * FDIM; i += 256) {           // X_b, pad rows
        int n = i / FDIM, f = i - n * FDIM;
        xs[i] = (n < NV) ? X[((size_t)b * NV + n) * FDIM + f] : 0.f;
    }
    for (int i = tid; i < FDIM * NPAD; i += 256) {           // kernels_h, pad cols
        int f = i / NPAD, j = i - f * NPAD;
        ks[i] = (j < NV) ? kernels[((size_t)h * FDIM + f) * NV + j] : 0.f;
    }
    for (int i = tid; i < FDIM * UDIM; i += 256)
        fcs[i] = fc[(size_t)h * FDIM * UDIM + i];
    for (int i = tid; i < NPAD; i += 256)
        b1s[i] = (i < NV) ? bias1[h * NV + i] : 0.f;
    if (tid < UDIM) b2s[tid] = bias2[h * UDIM + tid];
    __syncthreads();

    const int tw = wave & 3;      // output f/u tile owned by this wave
    const int kh = wave >> 2;     // K-half for split GEMMs

    for (int mb = 0; mb < NPAD / 16; ++mb) {
        const int m0 = mb * 16;

        // ---- stage AWT row-block (16 x 208) --------------------------------
        for (int i = tid; i < 16 * NPAD; i += 256) {
            int rr = i / NPAD, cc = i - rr * NPAD;
            awt[i] = AWT[((size_t)h * NPAD + m0 + rr) * NPAD + cc];
        }
        __syncthreads();

        // ---- feat = AWT_block (16x208) @ X_b (208x64), K-split 8 waves -----
        {
            const int f0 = tw * 16;
            v8f c = {};
            c = gemm_tile(awt, NPAD, xs, FDIM, f0, kh * KHALF, (kh + 1) * KHALF, c);
            if (kh == 1)   // upper-half partial -> scratch (ds region is dead)
                for (int r = 0; r < 8; ++r)
                    ds[(r + 8 * hi) * FDIM + f0 + lo] = c[r];
            __syncthreads();
            if (kh == 0)
                for (int r = 0; r < 8; ++r)
                    fs[(r + 8 * hi) * FDIM + f0 + lo] =
                        c[r] + ds[(r + 8 * hi) * FDIM + f0 + lo];
        }
        // ---- stage adjacency row-block (16 x 208) into awt (now dead) ------
        __syncthreads();
        for (int i = tid; i < 16 * NPAD; i += 256) {
            int rr = i / NPAD, cc = i - rr * NPAD;
            int m = m0 + rr;
            awt[i] = (m < NV && cc < NV) ? A[m * NV + cc] : 0.f;
        }
        __syncthreads();

        // ---- dense = feat (16x64) @ kernels (64x208) + bias1, masked -------
        for (int jt = wave; jt < NPAD / 16; jt += 8) {
            const int j0 = jt * 16;
            v8f c = {};
            c = gemm_tile(fs, FDIM, ks, NPAD, j0, 0, FDIM, c);
            const int j = j0 + lo;
            for (int r = 0; r < 8; ++r) {
                const int rr = r + 8 * hi;
                float adj = awt[rr * NPAD + j];
                ds[rr * NPAD + j] = c[r] + b1s[j] - NEGF * (1.f - adj);
            }
        }
        __syncthreads();

        // ---- row softmax over 208 cols (masked entries underflow to 0) -----
        {
            const int row = tid >> 4, cc = tid & 15;
            float mx = -3.0e38f;
            for (int j = cc; j < NPAD; j += 16)
                mx = fmaxf(mx, ds[row * NPAD + j]);
            for (int off = 8; off; off >>= 1)
                mx = fmaxf(mx, __shfl_xor(mx, off, 16));
            float sum = 0.f;
            for (int j = cc; j < NPAD; j += 16) {
                float e = __expf(ds[row * NPAD + j] - mx);
                ds[row * NPAD + j] = e;
                sum += e;
            }
            for (int off = 8; off; off >>= 1)
                sum += __shfl_xor(sum, off, 16);
            const float inv = 1.f / sum;
            for (int j = cc; j < NPAD; j += 16)
                ds[row * NPAD + j] *= inv;
        }
        __syncthreads();

        // ---- node = mask (16x208) @ X_b (208x64), K-split 8 waves ----------
        {
            const int f0 = tw * 16;
            v8f c = {};
            c = gemm_tile(ds, NPAD, xs, FDIM, f0, kh * KHALF, (kh + 1) * KHALF, c);
            if (kh == 1)   // partial -> awt region (adjacency block is dead)
                for (int r = 0; r < 8; ++r)
                    awt[(r + 8 * hi) * FDIM + f0 + lo] = c[r];
            __syncthreads();
            if (kh == 0)
                for (int r = 0; r < 8; ++r)
                    ns[(r + 8 * hi) * FDIM + f0 + lo] =
                        c[r] + awt[(r + 8 * hi) * FDIM + f0 + lo];
        }
        __syncthreads();

        // ---- out = node (16x64) @ fc (64x64) + bias2 -> global -------------
        if (wave < 4) {
            const int u0 = tw * 16;
            const float bb = b2s[u0 + lo];
            v8f c;
            for (int r = 0; r < 8; ++r) c[r] = bb;
            c = gemm_tile(ns, FDIM, fcs, UDIM, u0, 0, FDIM, c);
            for (int r = 0; r < 8; ++r) {
                const int m = m0 + r + 8 * hi;
                if (m < NV)
                    out[((size_t)b * NV + m) * OUTC + h * UDIM + u0 + lo] = c[r];
            }
        }

        // ---- last head also emits its mask rows into concat columns --------
        if (h == HNUM - 1) {
            for (int i = tid; i < 16 * NV; i += 256) {
                int rm = i / NV, j = i - rm * NV;
                int m = m0 + rm;
                if (m < NV)
                    out[((size_t)b * NV + m) * OUTC + HNUM * UDIM + j] =
                        ds[rm * NPAD + j];
            }
        }
        __syncthreads();
    }
}

// ---------------------------------------------------------------------------
extern "C" void kernel_launch(void* const* d_in, const int* in_sizes, int n_in,
                              void* d_out, int out_size, void* d_ws, size_t ws_size,
                              hipStream_t stream) {
    const float* X      = (const float*)d_in[0];
    const float* A      = (const float*)d_in[1];
    const float* kernls = (const float*)d_in[2];
    const float* params = (const float*)d_in[3];
    const float* fc     = (const float*)d_in[4];
    const float* bias1  = (const float*)d_in[5];
    const float* bias2  = (const float*)d_in[6];
    float* out = (float*)d_out;
    float* AWT = (float*)d_ws;                 // H*208*208 floats = 692 KB

    const int B = in_sizes[0] / (NV * FDIM);   // 512

    const int total = HNUM * NPAD * NPAD;
    prep_awt<<<(total + 255) / 256, 256, 0, stream>>>(A, params, AWT);

    (void)hipFuncSetAttribute((const void*)gc_main,
                              hipFuncAttributeMaxDynamicSharedMemorySize,
                              (int)SMEM_BYTES);
    dim3 grid(B, HNUM);
    gc_main<<<grid, 256, SMEM_BYTES, stream>>>(X, A, kernls, fc, bias1, bias2,
                                               AWT, out);
}